// Loss_67637144977961
// MI455X (gfx1250) — compile-verified
//
#include <hip/hip_runtime.h>

namespace {

constexpr int kS    = 1024;
constexpr int kB    = 2;
constexpr int kV    = 32000;
constexpr int kT    = kS - 1;               // 1023 token positions
constexpr int kRows = kB * kT;              // 2046 rows
constexpr int kF4   = kV / 4;               // 8000 float4 per row
constexpr int kBlk  = 256;                  // 8 wave32s
constexpr int kIters = (kF4 + kBlk - 1) / kBlk;  // 32
constexpr int kStages = 4;                  // async pipeline depth

constexpr float kGamma  = 1.0f;
constexpr float kLam    = 0.95f;
constexpr float kClipV  = 0.2f;
constexpr float kClip   = 0.2f;
constexpr float kVfCoef = 0.1f;

// ---- CDNA5 async global -> LDS copy (ASYNCcnt path, GVS addressing) -------
// saddr = uniform 64-bit row base (SGPR pair), vaddr = per-lane 32-bit byte
// offset, vdst = per-lane wave-relative LDS byte offset.
__device__ __forceinline__ void async_copy_b128(unsigned lds_off, unsigned goff,
                                                unsigned long long sbase) {
  asm volatile("global_load_async_to_lds_b128 %0, %1, %2"
               :: "v"(lds_off), "v"(goff), "s"(sbase)
               : "memory");
}
__device__ __forceinline__ void wait_async_le6() {
  asm volatile("s_wait_asynccnt 0x6" ::: "memory");
}
__device__ __forceinline__ void wait_async_le4() {
  asm volatile("s_wait_asynccnt 0x4" ::: "memory");
}
__device__ __forceinline__ void wait_async_le2() {
  asm volatile("s_wait_asynccnt 0x2" ::: "memory");
}
__device__ __forceinline__ void wait_async_0() {
  asm volatile("s_wait_asynccnt 0x0" ::: "memory");
}

// Online softmax accumulators (rescale only when a new max appears: 1 exp per
// element on the common path).
__device__ __forceinline__ void onl3(float x, float& m, float& s, float& d) {
  if (x > m) {
    float e = __expf(m - x);           // exp(-inf)=0 handles first element
    s = fmaf(s, e, 1.0f);
    d = fmaf(d, e, x);
    m = x;
  } else {
    float e = __expf(x - m);
    s += e;
    d = fmaf(e, x, d);
  }
}
__device__ __forceinline__ void onl2(float x, float& m, float& s) {
  if (x > m) {
    s = fmaf(s, __expf(m - x), 1.0f);
    m = x;
  } else {
    s += __expf(x - m);
  }
}

// ---------------------------------------------------------------------------
// Phase 1: per-row log-softmax stats for logits and old_logits.
// One block per (b,t) row; 4-stage async LDS pipeline, wave-private waits.
// ---------------------------------------------------------------------------
__global__ __launch_bounds__(kBlk)
void row_softmax_kernel(const float* __restrict__ logits,
                        const float* __restrict__ oldl,
                        const int*   __restrict__ ids,
                        float* __restrict__ out_lp,
                        float* __restrict__ out_olp,
                        float* __restrict__ out_ent) {
  __shared__ float4 bufL[kStages][kBlk];
  __shared__ float4 bufO[kStages][kBlk];
  __shared__ float  red[8][7];

  const int r = blockIdx.x;            // r = b*kT + t
  const int b = r / kT;
  const int t = r % kT;
  const long long rowoff = ((long long)t * kB + b) * kV;
  const unsigned long long gL = (unsigned long long)(logits + rowoff);
  const unsigned long long gO = (unsigned long long)(oldl   + rowoff);
  const int id = ids[b * kS + t + 1];

  const int tid = threadIdx.x;
  constexpr unsigned kStageBytes = (unsigned)(kBlk * 16);
  const unsigned baseL = (unsigned)(size_t)&bufL[0][0] + 16u * (unsigned)tid;
  const unsigned baseO = (unsigned)(size_t)&bufO[0][0] + 16u * (unsigned)tid;

  auto issue_tile = [&](int jt) {
    int f  = jt * kBlk + tid;
    int cf = f < kF4 ? f : kF4 - 1;                    // clamp (masked in compute)
    unsigned goff = (unsigned)cf * 16u;                // <= 127984, fits u32
    unsigned st   = (unsigned)(jt & (kStages - 1)) * kStageBytes;
    async_copy_b128(baseL + st, goff, gL);
    async_copy_b128(baseO + st, goff, gO);
  };

  // prologue: fill 3 of 4 stages (6 async ops in flight)
  issue_tile(0);
  issue_tile(1);
  issue_tile(2);

  float m  = -__builtin_inff(), s  = 0.0f, d = 0.0f;   // logits: max, sumexp, sum exp*x
  float m2 = -__builtin_inff(), s2 = 0.0f;             // old:    max, sumexp
  float tokL = 0.0f, tokO = 0.0f;                      // token logit pickup

  for (int j = 0; j < kIters; ++j) {
    if (j + 3 < kIters) {
      issue_tile(j + 3);               // 8 ops in flight
      wait_async_le6();                // oldest tile (j) done, 3 tiles in flight
    } else if (j + 2 < kIters) {
      wait_async_le4();
    } else if (j + 1 < kIters) {
      wait_async_le2();
    } else {
      wait_async_0();
    }

    const float4 Lv = bufL[j & (kStages - 1)][tid];    // ds_load_b128
    const float4 Ov = bufO[j & (kStages - 1)][tid];

    const int f = j * kBlk + tid;
    if (f < kF4) {
      onl3(Lv.x, m, s, d); onl3(Lv.y, m, s, d);
      onl3(Lv.z, m, s, d); onl3(Lv.w, m, s, d);
      onl2(Ov.x, m2, s2);  onl2(Ov.y, m2, s2);
      onl2(Ov.z, m2, s2);  onl2(Ov.w, m2, s2);
      const unsigned di = (unsigned)(id - 4 * f);
      if (di < 4u) {
        tokL += (di == 0) ? Lv.x : (di == 1) ? Lv.y : (di == 2) ? Lv.z : Lv.w;
        tokO += (di == 0) ? Ov.x : (di == 1) ? Ov.y : (di == 2) ? Ov.z : Ov.w;
      }
    }
  }

  // wave32 reduction (explicit width 32 — CDNA5 is wave32, not wave64)
  #pragma unroll
  for (int o = 16; o > 0; o >>= 1) {
    float mo = __shfl_xor(m, o, 32);
    float so = __shfl_xor(s, o, 32);
    float dd = __shfl_xor(d, o, 32);
    float mn = fmaxf(m, mo);
    float e0 = __expf(m - mn), e1 = __expf(mo - mn);
    s = s * e0 + so * e1;
    d = d * e0 + dd * e1;
    m = mn;
    float m2o = __shfl_xor(m2, o, 32);
    float s2o = __shfl_xor(s2, o, 32);
    float mn2 = fmaxf(m2, m2o);
    s2 = s2 * __expf(m2 - mn2) + s2o * __expf(m2o - mn2);
    m2 = mn2;
    tokL += __shfl_xor(tokL, o, 32);
    tokO += __shfl_xor(tokO, o, 32);
  }

  const int wv = tid >> 5;
  if ((tid & 31) == 0) {
    red[wv][0] = m;  red[wv][1] = s;  red[wv][2] = d;
    red[wv][3] = m2; red[wv][4] = s2;
    red[wv][5] = tokL; red[wv][6] = tokO;
  }
  __syncthreads();

  if (tid == 0) {
    float M = red[0][0], Sx = red[0][1], D = red[0][2];
    float M2 = red[0][3], S2 = red[0][4];
    float TL = red[0][5], TO = red[0][6];
    #pragma unroll
    for (int w = 1; w < kBlk / 32; ++w) {
      float mw = red[w][0], sw = red[w][1], dw = red[w][2];
      float mn = fmaxf(M, mw);
      float e0 = __expf(M - mn), e1 = __expf(mw - mn);
      Sx = Sx * e0 + sw * e1;
      D  = D  * e0 + dw * e1;
      M = mn;
      float m2w = red[w][3], s2w = red[w][4];
      float mn2 = fmaxf(M2, m2w);
      S2 = S2 * __expf(M2 - mn2) + s2w * __expf(m2w - mn2);
      M2 = mn2;
      TL += red[w][5];
      TO += red[w][6];
    }
    const float lse  = M  + __logf(Sx);
    const float lse2 = M2 + __logf(S2);
    out_lp[r]  = TL - lse;             // token logprob
    out_olp[r] = TO - lse2;            // old token logprob
    out_ent[r] = lse - D / Sx;         // logsumexp - sum(p*x)
  }
}

// ---------------------------------------------------------------------------
// Phase 2: GAE suffix scan (log-step), whitening, losses. One 1024-thread block.
// ---------------------------------------------------------------------------
__device__ __forceinline__ float4 block_sum4(float4 v, float4* sm4) {
  const int tid = threadIdx.x;
  sm4[tid] = v;
  __syncthreads();
  #pragma unroll
  for (int off = 512; off > 0; off >>= 1) {
    if (tid < off) {
      float4 o = sm4[tid + off];
      float4 a = sm4[tid];
      a.x += o.x; a.y += o.y; a.z += o.z; a.w += o.w;
      sm4[tid] = a;
    }
    __syncthreads();
  }
  float4 r = sm4[0];
  __syncthreads();
  return r;
}

__global__ __launch_bounds__(1024)
void ppo_loss_kernel(const float* __restrict__ values,
                     const float* __restrict__ vpreds,
                     const float* __restrict__ rewards,
                     const int*   __restrict__ mask,
                     const float* __restrict__ lp_arr,
                     const float* __restrict__ olp_arr,
                     const float* __restrict__ ent_arr,
                     float* __restrict__ out) {
  __shared__ float  adv[kB][kT + 1];
  __shared__ float4 sm4[1024];

  const int t = threadIdx.x;
  const bool act = t < kT;

  float mf[kB]  = {0.f, 0.f};
  float val[kB] = {0.f, 0.f};
  float vp[kB]  = {0.f, 0.f};

  if (act) {
    #pragma unroll
    for (int b = 0; b < kB; ++b) {
      float mfb = (float)mask[b * kT + t];
      float vb  = values[t * kB + b] * mfb;              // masked value
      float nv  = (t < kT - 1)
                    ? values[(t + 1) * kB + b] * (float)mask[b * kT + t + 1]
                    : 0.0f;
      mf[b]  = mfb;
      val[b] = vb;
      vp[b]  = vpreds[t * kB + b];
      adv[b][t] = rewards[b * kT + t] * mfb + kGamma * nv - vb;  // delta
    }
  }
  __syncthreads();

  // Suffix scan: adv[t] = sum_{k>=t} c^(k-t) delta[k], c = gamma*lambda.
  // Constant-coefficient recurrence -> log-step Hillis-Steele (c^512 ~ 4e-12,
  // well inside fp32 normal range).
  float c = kGamma * kLam;
  for (int o = 1; o < kT; o <<= 1) {
    float t0 = 0.f, t1 = 0.f;
    if (act && t + o < kT) { t0 = adv[0][t + o]; t1 = adv[1][t + o]; }
    __syncthreads();
    if (act) { adv[0][t] += c * t0; adv[1][t] += c * t1; }
    __syncthreads();
    c *= c;
  }

  float a[kB] = {0.f, 0.f};
  if (act) { a[0] = adv[0][t]; a[1] = adv[1][t]; }

  // Reduction 1: mask sum + masked advantage sum
  float4 r1 = block_sum4(make_float4(mf[0] + mf[1],
                                     a[0] * mf[0] + a[1] * mf[1], 0.f, 0.f), sm4);
  const float msum = r1.x;
  const float mean = r1.y / msum;

  // Reduction 2: variance numerator + vf + entropy + kl (mean-independent parts)
  float varn = 0.f, vf_acc = 0.f, ent_acc = 0.f, kl_acc = 0.f;
  if (act) {
    #pragma unroll
    for (int b = 0; b < kB; ++b) {
      const int rr = b * kT + t;
      const float dv = a[b] - mean;
      varn += mf[b] * dv * dv;

      const float ret = a[b] + val[b];
      const float vpc = fminf(fmaxf(vp[b], val[b] - kClipV), val[b] + kClipV);
      const float v1 = (vp[b] - ret) * (vp[b] - ret);
      const float v2 = (vpc  - ret) * (vpc  - ret);
      vf_acc += fmaxf(v1, v2) * mf[b];

      ent_acc += ent_arr[rr] * mf[b];
      const float dlp = lp_arr[rr] - olp_arr[rr];
      kl_acc += dlp * dlp * mf[b];
    }
  }
  float4 r2 = block_sum4(make_float4(varn, vf_acc, ent_acc, kl_acc), sm4);
  float var = (r2.x / msum) * (msum / (msum - 1.0f));   // unbiased
  const float inv = rsqrtf(var + 1e-8f);

  // Reduction 3: policy-gradient loss (needs whitened advantages)
  float pg_acc = 0.f;
  if (act) {
    #pragma unroll
    for (int b = 0; b < kB; ++b) {
      const int rr = b * kT + t;
      const float ratio = __expf(lp_arr[rr] - olp_arr[rr]);
      const float aw = (a[b] - mean) * inv;
      const float p1 = -aw * ratio;
      const float p2 = -aw * fminf(fmaxf(ratio, 1.0f - kClip), 1.0f + kClip);
      pg_acc += fmaxf(p1, p2) * mf[b];
    }
  }
  float4 r3 = block_sum4(make_float4(pg_acc, 0.f, 0.f, 0.f), sm4);

  if (t == 0) {
    const float vf_loss = 0.5f * r2.y / msum;
    const float pg_loss = r3.x / msum;
    out[0] = pg_loss + kVfCoef * vf_loss;   // loss
    out[1] = pg_loss;                       // pg_loss
    out[2] = vf_loss;                       // vf_loss
    out[3] = r2.z / msum;                   // entropy
    out[4] = 0.5f * r2.w / msum;            // approxkl
  }
}

}  // namespace

extern "C" void kernel_launch(void* const* d_in, const int* in_sizes, int n_in,
                              void* d_out, int out_size, void* d_ws, size_t ws_size,
                              hipStream_t stream) {
  (void)in_sizes; (void)n_in; (void)out_size; (void)ws_size;
  const int*   ids     = (const int*)d_in[0];     // input_ids   (B,S) int32
  const float* logits  = (const float*)d_in[1];   // sharded_logits (S,B,V) f32
  const float* oldl    = (const float*)d_in[2];   // old_logits     (S,B,V) f32
  const float* values  = (const float*)d_in[3];   // values  (S,B,1) f32
  const float* vpreds  = (const float*)d_in[4];   // vpreds  (S,B,1) f32
  const float* rewards = (const float*)d_in[5];   // rewards (B,S-1) f32
  const int*   mask    = (const int*)d_in[6];     // mask    (B,S-1) int32
  float* out = (float*)d_out;                     // 5 scalars

  float* ws     = (float*)d_ws;                   // 3 * kRows floats (~24 KB)
  float* ws_lp  = ws;
  float* ws_olp = ws + kRows;
  float* ws_ent = ws + 2 * kRows;

  row_softmax_kernel<<<kRows, kBlk, 0, stream>>>(logits, oldl, ids,
                                                 ws_lp, ws_olp, ws_ent);
  ppo_loss_kernel<<<1, 1024, 0, stream>>>(values, vpreds, rewards, mask,
                                          ws_lp, ws_olp, ws_ent, out);
}